// FMatrixLayer_52226802319912
// MI455X (gfx1250) — compile-verified
//
#include <hip/hip_runtime.h>

// FMatrix bound-propagation scan for MI455X (gfx1250, wave32).
// One thread per batch element; column-major traversal of the triangular
// recurrence so the flat index k advances by +1 per cell (streaming I/O).
// Per-thread column state lives in LDS (126 floats/thread, padded to 127).
// Global x-loads and LDS 'left'-loads are software-pipelined one iteration
// ahead; WGP-scope global_prefetch_b8 (inline asm, scope 0) stages the next
// column's cache lines into L0 while the current column computes.

#define NDIM      128
#define IN_DIM    8001              // (N-1)*(N-2)/2
#define BATCH     2048
#define BLOCK     64                // 2 wave32 waves per block
#define NBLK      (BATCH / BLOCK)   // 32 blocks
#define SCALE     129.0f
#define INV_SCALE (1.0f / 129.0f)

__device__ __forceinline__ int clamp_k(int k) {
    return k < IN_DIM ? k : (IN_DIM - 1);   // wave-uniform -> SALU s_min
}

// WGP-scope prefetch: unannotated scope == SCOPE 0 == "pull into all cache
// levels on miss" (the builtin only ever emits SE/SYS scope, which stop at GL2).
__device__ __forceinline__ void prefetch_l0(const float* p) {
    asm volatile("global_prefetch_b8 %0, off" : : "v"(p));
}

__global__ __launch_bounds__(BLOCK) void fmatrix_scan_kernel(
    const float* __restrict__ x, float* __restrict__ out,
    float* __restrict__ loss_part)
{
    // slot (i-2) holds final(i, j-1) while processing column j.
    // +1 padding row so the pipelined 'left' preload at i==127 stays in bounds.
    __shared__ float sbuf[(NDIM - 1) * BLOCK];
    __shared__ float sred[BLOCK];

    const int tid = threadIdx.x;
    const int b   = blockIdx.x * BLOCK + tid;

    const float* __restrict__ xp = x   + (size_t)b * IN_DIM;
    float*       __restrict__ op = out + (size_t)b * IN_DIM;

    float loss = 0.0f;
    int   k    = 0;

    // Pipelined x value (already scaled).
    float xv = SCALE * xp[0];

    // ---- column j = 0 : rows i = 2..127 (special j==0 update) ----
    float prev0 = 1.0f;                       // final(1,0) subdiag seed
    for (int i = 2; i < NDIM; ++i) {
        float xnext = xp[clamp_k(k + 1)];     // issue next load early
        prefetch_l0(xp + clamp_k(k + 128));   // warm-up window for cols 1..2
        float curr  = xv;
        float lower = fmaxf(prev0 - 1.0f, 0.0f);
        float vlo   = fmaxf(lower - curr, 0.0f);
        float vhi   = fmaxf(curr - prev0, 0.0f);
        float nv    = curr + vlo - vhi;
        loss += vlo + vhi;
        __builtin_nontemporal_store(nv * INV_SCALE, op + k);
        sbuf[(i - 2) * BLOCK + tid] = nv;
        prev0 = nv;
        xv = SCALE * xnext;
        ++k;
    }

    // ---- columns j = 1..125 : rows i = j+2..127 (general update) ----
    for (int j = 1; j <= NDIM - 3; ++j) {
        // Prefetch the NEXT column's span (<= 126 floats = <= 4+1 lines) while
        // this column computes; lead time ~ (126-j)*15 cycles >> HBM latency.
        {
            int kn = k + (NDIM - 2 - j);      // next column's first flat index
            prefetch_l0(xp + clamp_k(kn));
            prefetch_l0(xp + clamp_k(kn + 32));
            prefetch_l0(xp + clamp_k(kn + 64));
            prefetch_l0(xp + clamp_k(kn + 96));
        }

        float up   = (float)(j + 1);               // final(j+1, j) subdiag
        float dg   = sbuf[(j - 1) * BLOCK + tid];  // final(j+1, j-1)
        float left = sbuf[j * BLOCK + tid];        // final(j+2, j-1), slot j
        for (int i = j + 2; i < NDIM; ++i) {
            // Pipelined loads for the NEXT iteration (slot i-1 is written in the
            // next iteration, so this read still sees column j-1).
            float left_n = sbuf[(i - 1) * BLOCK + tid];
            float xnext  = xp[clamp_k(k + 1)];

            float curr = xv;
            float l3   = left + up - dg - 1.0f;
            float fl   = fmaxf(fmaxf(left, fmaxf(up - 1.0f, l3)), 0.0f);
            float fu   = fmaxf(fminf(up, left + up - dg), fl);
            float vlo  = fmaxf(fl - curr, 0.0f);
            float vhi  = fmaxf(curr - fu, 0.0f);
            float nv   = curr + vlo - vhi;
            loss += vlo + vhi;
            __builtin_nontemporal_store(nv * INV_SCALE, op + k);
            sbuf[(i - 2) * BLOCK + tid] = nv;      // becomes final(i, j)

            dg   = left;                           // old final(i, j-1)
            up   = nv;                             // final(i, j)
            left = left_n;
            xv   = SCALE * xnext;
            ++k;
        }
    }

    // ---- deterministic per-block loss reduction ----
    sred[tid] = loss;
    __syncthreads();
    for (int s = BLOCK / 2; s > 0; s >>= 1) {
        if (tid < s) sred[tid] += sred[tid + s];
        __syncthreads();
    }
    if (tid == 0) loss_part[blockIdx.x] = sred[0];
}

__global__ void fmatrix_loss_finalize(const float* __restrict__ loss_part,
                                      float* __restrict__ out)
{
    if (blockIdx.x == 0 && threadIdx.x == 0) {
        float s = 0.0f;
        for (int i = 0; i < NBLK; ++i) s += loss_part[i];
        out[(size_t)BATCH * IN_DIM] = s;   // tuple output #2: total_loss
    }
}

extern "C" void kernel_launch(void* const* d_in, const int* in_sizes, int n_in,
                              void* d_out, int out_size, void* d_ws, size_t ws_size,
                              hipStream_t stream) {
    const float* x       = (const float*)d_in[0];
    float*       out     = (float*)d_out;
    float*       loss_ws = (float*)d_ws;     // NBLK floats of scratch

    fmatrix_scan_kernel<<<NBLK, BLOCK, 0, stream>>>(x, out, loss_ws);
    fmatrix_loss_finalize<<<1, 32, 0, stream>>>(loss_ws, out);

    (void)in_sizes; (void)n_in; (void)out_size; (void)ws_size;
}